// SGCLDGA_73134703116394
// MI455X (gfx1250) — compile-verified
//
#include <hip/hip_runtime.h>
#include <math.h>

// ---------------- problem constants (match reference) ----------------
#define N_G      50000
#define N_D      50000
#define DIM      64
#define RANK     32
#define NNZ_C    1600000
#define BATCH    2048
#define TEMP     0.2f
#define LAMBDA1  0.2f
#define LAMBDA2  1e-7f
#define DROPP    0.1f

typedef __attribute__((ext_vector_type(2))) float v2f;
typedef __attribute__((ext_vector_type(8))) float v8f;

// D = A(16x4 f32) * B(4x16 f32) + C  (CDNA5 V_WMMA_F32_16X16X4_F32)
__device__ __forceinline__ v8f wmma_k4(v2f a, v2f b, v8f c) {
    return __builtin_amdgcn_wmma_f32_16x16x4_f32(
        /*neg_a=*/false, a, /*neg_b=*/false, b,
        /*c_mod=*/(short)0, c, /*reuse_a=*/false, /*reuse_b=*/false);
}

// ------------------------------------------------------------------
// 1) COO SpMM with sparse dropout: Z[seg[e]] += w_e * Ein[gat[e]]
//    thread = (edge, 4-wide d chunk); 16 threads per edge.
// ------------------------------------------------------------------
__global__ void spmm_scatter(const int* __restrict__ seg,
                             const int* __restrict__ gat,
                             const float* __restrict__ vals,
                             const float* __restrict__ dropu,
                             const float* __restrict__ Ein,
                             float* __restrict__ Zout, int nnz) {
    long long t = (long long)blockIdx.x * blockDim.x + threadIdx.x;
    int e  = (int)(t >> 4);
    int d  = ((int)t & 15) << 2;
    if (e >= nnz) return;
    float keep = (dropu[e] > DROPP) ? (1.0f / (1.0f - DROPP)) : 0.0f;
    float w = vals[e] * keep;
    if (w == 0.0f) return;
    const float4 s = *(const float4*)(Ein + (long long)gat[e] * DIM + d);
    float* dst = Zout + (long long)seg[e] * DIM + d;
    atomicAdd(dst + 0, w * s.x);
    atomicAdd(dst + 1, w * s.y);
    atomicAdd(dst + 2, w * s.z);
    atomicAdd(dst + 3, w * s.w);
}

// ------------------------------------------------------------------
// 2) acc[i] += a[i] + b[i]   (E_sum built in-place over Z(layer1))
// ------------------------------------------------------------------
__global__ void add3_inplace(float* __restrict__ acc,
                             const float* __restrict__ a,
                             const float* __restrict__ b, int n) {
    int i = blockIdx.x * blockDim.x + threadIdx.x;
    if (i < n) acc[i] += a[i] + b[i];
}

// ------------------------------------------------------------------
// 3) T[32,64] = wt[32,N] @ (E0 + Z)[N,64]   (tall reduction, atomics)
//    thread (rb = tid>>6 in 0..3, d = tid&63) owns rows {rb, rb+4, ...}
// ------------------------------------------------------------------
#define LR_CHUNK 256
__global__ void lowrank_reduce(const float* __restrict__ wt,
                               const float* __restrict__ E0,
                               const float* __restrict__ Z,
                               float* __restrict__ T, int N) {
    int d  = threadIdx.x & 63;
    int rb = threadIdx.x >> 6;
    float acc[8];
#pragma unroll
    for (int k = 0; k < 8; ++k) acc[k] = 0.0f;
    int j0 = blockIdx.x * LR_CHUNK;
    int j1 = min(j0 + LR_CHUNK, N);
    for (int j = j0; j < j1; ++j) {
        float e = E0[(long long)j * DIM + d] + Z[(long long)j * DIM + d];
#pragma unroll
        for (int k = 0; k < 8; ++k)
            acc[k] += wt[(long long)(rb + 4 * k) * N + j] * e;
    }
#pragma unroll
    for (int k = 0; k < 8; ++k) atomicAdd(&T[(rb + 4 * k) * DIM + d], acc[k]);
}

// ------------------------------------------------------------------
// 4) G[N,64] = E0[N,64] + M[N,32] @ T[32,64]   via WMMA f32 16x16x4
//    wave -> one 16x16 tile; block(256)=8 waves = 2 row tiles x 4 col tiles
// ------------------------------------------------------------------
__global__ void lowrank_expand(const float* __restrict__ E0,
                               const float* __restrict__ M,
                               const float* __restrict__ T,
                               float* __restrict__ G, int N) {
    int wave = threadIdx.x >> 5;
    int lane = threadIdx.x & 31;
    int ct   = wave & 3;
    int rt   = (blockIdx.x << 1) + (wave >> 2);
    int row0 = rt * 16;
    if (row0 >= N) return;
    int n0  = ct * 16;
    int l16 = lane & 15;
    int hi  = lane >> 4;

    v8f c;
#pragma unroll
    for (int v = 0; v < 8; ++v) {
        int r = row0 + v + hi * 8;
        c[v] = E0[(long long)r * DIM + n0 + l16];
    }
    long long arow = (long long)(row0 + l16) * RANK;
#pragma unroll
    for (int kk = 0; kk < RANK; kk += 4) {
        int k2 = kk + 2 * hi;
        float2 av = *(const float2*)(M + arow + k2);
        v2f a; a.x = av.x; a.y = av.y;
        v2f b; b.x = T[(k2 + 0) * DIM + n0 + l16];
               b.y = T[(k2 + 1) * DIM + n0 + l16];
        c = wmma_k4(a, b, c);
    }
#pragma unroll
    for (int v = 0; v < 8; ++v) {
        int r = row0 + v + hi * 8;
        G[(long long)r * DIM + n0 + l16] = c[v];
    }
}

// ------------------------------------------------------------------
// 5) Fused contrastive negatives: rowsum[b] += sum_j exp(dot(Q_b,E_j)/T)
//    Q_b = G[ids[b]].  wave -> 16x16 logits tile, 16 WMMAs over K=64,
//    exp + 16-lane butterfly row reduction + 16 atomics per tile.
// ------------------------------------------------------------------
__global__ void neg_score_gemm(const int* __restrict__ ids,
                               const float* __restrict__ G,
                               const float* __restrict__ Es,
                               float* __restrict__ rowsum, int Ncols) {
    int wave = threadIdx.x >> 5;
    int lane = threadIdx.x & 31;
    int ct   = blockIdx.x * 8 + wave;
    if (ct * 16 >= Ncols) return;
    int rt  = blockIdx.y;
    int l16 = lane & 15;
    int hi  = lane >> 4;

    long long abase = (long long)ids[rt * 16 + l16] * DIM;
    long long bbase = (long long)(ct * 16 + l16) * DIM;

    v8f c = {};
#pragma unroll
    for (int kk = 0; kk < DIM; kk += 4) {
        int k2 = kk + 2 * hi;
        float2 av = *(const float2*)(G  + abase + k2);
        float2 bv = *(const float2*)(Es + bbase + k2);
        v2f a; a.x = av.x; a.y = av.y;
        v2f b; b.x = bv.x; b.y = bv.y;
        c = wmma_k4(a, b, c);
    }
    const float invT = 1.0f / TEMP;
#pragma unroll
    for (int v = 0; v < 8; ++v) c[v] = expf(c[v] * invT);
    // reduce across the 16 N-lanes of each half-wave
#pragma unroll
    for (int m = 1; m < 16; m <<= 1) {
#pragma unroll
        for (int v = 0; v < 8; ++v) c[v] += __shfl_xor(c[v], m, 32);
    }
    if (l16 == 0) {
#pragma unroll
        for (int v = 0; v < 8; ++v)
            atomicAdd(&rowsum[rt * 16 + v + hi * 8], c[v]);
    }
}

// ------------------------------------------------------------------
// 6) per-sample positive score + BPR loss
//    accum[0]=sum pos_score, accum[1]=sum bpr
// ------------------------------------------------------------------
__global__ void pair_losses(const int* __restrict__ uids, const int* __restrict__ iids,
                            const int* __restrict__ pos,  const int* __restrict__ neg,
                            const float* __restrict__ Ggs, const float* __restrict__ Egs,
                            const float* __restrict__ Gds, const float* __restrict__ Eds,
                            float* __restrict__ accum, int B) {
    int b = blockIdx.x * blockDim.x + threadIdx.x;
    if (b >= B) return;
    long long u  = (long long)uids[b] * DIM;
    long long ii = (long long)iids[b] * DIM;
    long long p  = (long long)pos[b]  * DIM;
    long long ng = (long long)neg[b]  * DIM;
    float pg = 0.f, pd = 0.f, dp = 0.f, dn = 0.f;
    for (int k = 0; k < DIM; ++k) {
        float eg = Egs[u + k];
        pg += Ggs[u + k] * eg;
        pd += Gds[ii + k] * Eds[ii + k];
        dp += eg * Eds[p + k];
        dn += eg * Eds[ng + k];
    }
    const float invT = 1.0f / TEMP;
    float ps = fminf(fmaxf(pg * invT, -5.f), 5.f) + fminf(fmaxf(pd * invT, -5.f), 5.f);
    float diff = dp - dn;
    float lr = (diff > 0.f) ? log1pf(expf(-diff)) : (-diff + log1pf(expf(diff)));
    atomicAdd(&accum[0], ps);
    atomicAdd(&accum[1], lr);
}

// 7) sum of squares (L2 reg) -> accum[2]
__global__ void sq_reduce(const float* __restrict__ x, long long n,
                          float* __restrict__ out) {
    long long i = (long long)blockIdx.x * blockDim.x + threadIdx.x;
    long long stride = (long long)gridDim.x * blockDim.x;
    float s = 0.f;
    for (; i < n; i += stride) { float v = x[i]; s += v * v; }
#pragma unroll
    for (int m = 1; m < 32; m <<= 1) s += __shfl_xor(s, m, 32);
    if ((threadIdx.x & 31) == 0) atomicAdd(out, s);
}

// 8) log(rowsum + eps) sums -> accum[3], accum[4]
__global__ void log_rowsum(const float* __restrict__ rsg,
                           const float* __restrict__ rsd,
                           float* __restrict__ accum, int B) {
    int b = blockIdx.x * blockDim.x + threadIdx.x;
    if (b >= B) return;
    atomicAdd(&accum[3], logf(rsg[b] + 1e-8f));
    atomicAdd(&accum[4], logf(rsd[b] + 1e-8f));
}

// 9) final scalar assembly
__global__ void finalize(const float* __restrict__ accum, float* __restrict__ out) {
    const float invB = 1.0f / (float)BATCH;
    float pos  = accum[0] * invB;
    float lr   = accum[1] * invB;
    float reg  = LAMBDA2 * accum[2];
    float negs = accum[3] * invB + accum[4] * invB;
    float ls   = -pos + negs;
    out[0] = lr + LAMBDA1 * ls + reg;
    out[1] = lr;
    out[2] = LAMBDA1 * ls;
}

// ------------------------------------------------------------------
extern "C" void kernel_launch(void* const* d_in, const int* in_sizes, int n_in,
                              void* d_out, int out_size, void* d_ws, size_t ws_size,
                              hipStream_t stream) {
    const int*   uids   = (const int*)  d_in[0];
    const int*   iids   = (const int*)  d_in[1];
    const int*   pos    = (const int*)  d_in[2];
    const int*   neg    = (const int*)  d_in[3];
    const float* Eg0    = (const float*)d_in[4];
    const float* Ed0    = (const float*)d_in[5];
    const int*   rows   = (const int*)  d_in[6];
    const int*   cols   = (const int*)  d_in[7];
    const float* vals   = (const float*)d_in[8];
    const float* g_mul  = (const float*)d_in[9];   // [N_G, R]
    const float* v_mul  = (const float*)d_in[10];  // [N_D, R]
    const float* ut     = (const float*)d_in[11];  // [R, N_G]
    const float* vt     = (const float*)d_in[12];  // [R, N_D]
    const float* dropu  = (const float*)d_in[13];  // [L,2,NNZ]
    float* out = (float*)d_out;

    const long long NE = (long long)N_G * DIM;     // 3.2M floats per matrix
    float* ws   = (float*)d_ws;
    float* Zg0  = ws;                              // -> later G_g_sum
    float* Zd0  = ws + 1 * NE;                     // -> later G_d_sum
    float* Zg1  = ws + 2 * NE;                     // -> later E_g_sum
    float* Zd1  = ws + 3 * NE;                     // -> later E_d_sum
    float* Tg   = ws + 4 * NE;                     // [32,64]
    float* Td   = Tg + RANK * DIM;
    float* rsg  = Td + RANK * DIM;                 // rowsum_g [B]
    float* rsd  = rsg + BATCH;                     // rowsum_d [B]
    float* acc  = rsd + BATCH;                     // 8 scalar accumulators

    // zero Z buffers + small region (graph-capture-legal memset node)
    hipMemsetAsync(d_ws, 0, (size_t)(4 * NE + 2 * RANK * DIM + 2 * BATCH + 8) * 4, stream);

    // ---- layer 0 SpMM ----
    {
        long long threads = (long long)NNZ_C * 16;
        int blocks = (int)((threads + 255) / 256);
        spmm_scatter<<<blocks, 256, 0, stream>>>(rows, cols, vals, dropu + 0LL * NNZ_C, Ed0, Zg0, NNZ_C);
        spmm_scatter<<<blocks, 256, 0, stream>>>(cols, rows, vals, dropu + 1LL * NNZ_C, Eg0, Zd0, NNZ_C);
        // ---- layer 1 SpMM (inputs are layer-0 outputs) ----
        spmm_scatter<<<blocks, 256, 0, stream>>>(rows, cols, vals, dropu + 2LL * NNZ_C, Zd0, Zg1, NNZ_C);
        spmm_scatter<<<blocks, 256, 0, stream>>>(cols, rows, vals, dropu + 3LL * NNZ_C, Zg0, Zd1, NNZ_C);
    }

    // ---- E_sum = E0 + Z(l0) + Z(l1), in place over Z(l1) ----
    {
        int n = (int)NE;
        int blocks = (n + 255) / 256;
        add3_inplace<<<blocks, 256, 0, stream>>>(Zg1, Eg0, Zg0, n);
        add3_inplace<<<blocks, 256, 0, stream>>>(Zd1, Ed0, Zd0, n);
    }

    // ---- low-rank branch (linearity: G_sum = E0 + M @ (wt @ (E0 + Z_l0))) ----
    {
        int blocks = (N_D + LR_CHUNK - 1) / LR_CHUNK;
        lowrank_reduce<<<blocks, 256, 0, stream>>>(vt, Ed0, Zd0, Tg, N_D);
        lowrank_reduce<<<blocks, 256, 0, stream>>>(ut, Eg0, Zg0, Td, N_G);
        int eblocks = (N_G / 16 + 1) / 2;  // 2 row tiles per block
        lowrank_expand<<<eblocks + 1, 256, 0, stream>>>(Eg0, g_mul, Tg, Zg0, N_G); // G_g_sum -> Zg0
        lowrank_expand<<<eblocks + 1, 256, 0, stream>>>(Ed0, v_mul, Td, Zd0, N_D); // G_d_sum -> Zd0
    }

    // ---- fused negatives: rowsum = sum_j exp(S/T)  (WMMA GEMM) ----
    {
        dim3 grid((N_G / 16 + 7) / 8, BATCH / 16);
        neg_score_gemm<<<grid, 256, 0, stream>>>(uids, /*G_g_sum*/ Zg0, /*E_g_sum*/ Zg1, rsg, N_G);
        neg_score_gemm<<<grid, 256, 0, stream>>>(iids, /*G_d_sum*/ Zd0, /*E_d_sum*/ Zd1, rsd, N_D);
    }

    // ---- positives + BPR ----
    pair_losses<<<(BATCH + 255) / 256, 256, 0, stream>>>(uids, iids, pos, neg,
                                                         Zg0, Zg1, Zd0, Zd1, acc, BATCH);
    // ---- L2 reg ----
    sq_reduce<<<512, 256, 0, stream>>>(Eg0, NE, acc + 2);
    sq_reduce<<<512, 256, 0, stream>>>(Ed0, NE, acc + 2);
    // ---- log of row sums ----
    log_rowsum<<<(BATCH + 255) / 256, 256, 0, stream>>>(rsg, rsd, acc, BATCH);
    // ---- final 3 scalars ----
    finalize<<<1, 1, 0, stream>>>(acc, out);
}